// OTMatchContrastiveLoss_27900107555345
// MI455X (gfx1250) — compile-verified
//
#include <hip/hip_runtime.h>

typedef __attribute__((ext_vector_type(2))) float v2f;
typedef __attribute__((ext_vector_type(8))) float v8f;

#define N2048 2048
#define OT_MARGIN 0.7f
#define SINKHORN_ITERS 500
#define SK_BLOCKS 256                 // persistent kernel blocks (8 waves each)
#define SK_WAVES (SK_BLOCKS * 8)      // grid-stride over rows/cols

__device__ __forceinline__ v8f wmma_f32_k4(v2f a, v2f b, v8f c) {
  // D = A(16x4 f32) * B(4x16 f32) + C(16x16 f32)
  return __builtin_amdgcn_wmma_f32_16x16x4_f32(
      /*neg_a=*/false, a, /*neg_b=*/false, b,
      /*c_mod=*/(short)0, c, /*reuse_a=*/false, /*reuse_b=*/false);
}

// ---------------------------------------------------------------- init
__global__ __launch_bounds__(256) void ot_init_kernel(
    const float* __restrict__ qa, const float* __restrict__ ta,
    float* __restrict__ u, float* __restrict__ v,
    float* __restrict__ lmu, float* __restrict__ lnu,
    float* __restrict__ out, unsigned* __restrict__ bar) {
  int i = blockIdx.x * 256 + threadIdx.x;
  u[i] = 0.f;
  v[i] = 0.f;
  lmu[i] = __logf(qa[i]);
  lnu[i] = __logf(ta[i]);
  if (i == 0) {
    out[0] = 0.f;
    bar[0] = 0u;   // arrive counter (must start at 0)
    bar[1] = 0u;   // generation
  }
}

// ------------------------------------------------- column squared norms
// X is [D, N] row-major; sq[i] = sum_d X[d,i]^2 (coalesced over i)
__global__ __launch_bounds__(256) void ot_colsq_kernel(
    const float* __restrict__ X, float* __restrict__ sq) {
  int i = blockIdx.x * 256 + threadIdx.x;
  float s = 0.f;
  for (int d = 0; d < N2048; ++d) {
    float x = X[(size_t)d * N2048 + i];
    s = fmaf(x, x, s);
  }
  sq[i] = s;
}

// ------------------------------------------------------------- GEMM 1
// Z[i,j] = max(qsq[i] + tsq[j] - 2 * sum_d QF[d,i]*TF[d,j], 0); also ZT = Z^T.
// Block: 8 waves, each wave a 32x32 tile; block tile 64(M) x 128(N).
__global__ __launch_bounds__(256) void ot_gemm1_kernel(
    const float* __restrict__ QF, const float* __restrict__ TF,
    const float* __restrict__ qsq, const float* __restrict__ tsq,
    float* __restrict__ Z, float* __restrict__ ZT) {
  const int lane = threadIdx.x & 31;
  const int wave = threadIdx.x >> 5;
  const int i0 = blockIdx.x * 64 + (wave & 1) * 32;
  const int j0 = blockIdx.y * 128 + (wave >> 1) * 32;
  const int r16 = lane & 15;   // M (A) / N (B) index within 16
  const int g   = lane >> 4;   // K half: lanes 0-15 hold K=0,1; 16-31 hold K=2,3

  v8f acc00 = {}, acc01 = {}, acc10 = {}, acc11 = {};
  const float* Ab = QF + (size_t)(2 * g) * N2048;
  const float* Bb = TF + (size_t)(2 * g) * N2048;

  for (int d0 = 0; d0 < N2048; d0 += 4) {
    const float* Ap = Ab + (size_t)d0 * N2048;
    const float* Bp = Bb + (size_t)d0 * N2048;
    v2f a0, a1, b0, b1;
    a0.x = Ap[i0 + r16];            a0.y = Ap[N2048 + i0 + r16];
    a1.x = Ap[i0 + 16 + r16];       a1.y = Ap[N2048 + i0 + 16 + r16];
    b0.x = Bp[j0 + r16];            b0.y = Bp[N2048 + j0 + r16];
    b1.x = Bp[j0 + 16 + r16];       b1.y = Bp[N2048 + j0 + 16 + r16];
    acc00 = wmma_f32_k4(a0, b0, acc00);
    acc01 = wmma_f32_k4(a0, b1, acc01);
    acc10 = wmma_f32_k4(a1, b0, acc10);
    acc11 = wmma_f32_k4(a1, b1, acc11);
  }

  // C/D layout: VGPR r holds row r (lanes 0-15) / row 8+r (lanes 16-31), col = lane%16
#pragma unroll
  for (int r = 0; r < 8; ++r) {
    int ri0 = i0 + 8 * g + r;
    int ri1 = ri0 + 16;
    int cj0 = j0 + r16;
    int cj1 = cj0 + 16;
    float z00 = fmaxf(qsq[ri0] + tsq[cj0] - 2.f * acc00[r], 0.f);
    float z01 = fmaxf(qsq[ri0] + tsq[cj1] - 2.f * acc01[r], 0.f);
    float z10 = fmaxf(qsq[ri1] + tsq[cj0] - 2.f * acc10[r], 0.f);
    float z11 = fmaxf(qsq[ri1] + tsq[cj1] - 2.f * acc11[r], 0.f);
    Z[(size_t)ri0 * N2048 + cj0] = z00;
    Z[(size_t)ri0 * N2048 + cj1] = z01;
    Z[(size_t)ri1 * N2048 + cj0] = z10;
    Z[(size_t)ri1 * N2048 + cj1] = z11;
    ZT[(size_t)cj0 * N2048 + ri0] = z00;
    ZT[(size_t)cj1 * N2048 + ri0] = z01;
    ZT[(size_t)cj0 * N2048 + ri1] = z10;
    ZT[(size_t)cj1 * N2048 + ri1] = z11;
  }
}

// ------------------------------------------------------ grid-wide sync
__device__ __forceinline__ void ot_grid_sync(unsigned* cnt, unsigned* gen,
                                             unsigned nblocks) {
  __threadfence();   // release: make this block's stores visible at device scope
  __syncthreads();
  if (threadIdx.x == 0) {
    unsigned g = __hip_atomic_load(gen, __ATOMIC_RELAXED, __HIP_MEMORY_SCOPE_AGENT);
    unsigned arrived =
        __hip_atomic_fetch_add(cnt, 1u, __ATOMIC_ACQ_REL, __HIP_MEMORY_SCOPE_AGENT);
    if (arrived == nblocks - 1) {
      __hip_atomic_store(cnt, 0u, __ATOMIC_RELAXED, __HIP_MEMORY_SCOPE_AGENT);
      __hip_atomic_fetch_add(gen, 1u, __ATOMIC_RELEASE, __HIP_MEMORY_SCOPE_AGENT);
    } else {
      while (__hip_atomic_load(gen, __ATOMIC_RELAXED, __HIP_MEMORY_SCOPE_AGENT) == g)
        __builtin_amdgcn_s_sleep(2);
    }
  }
  __syncthreads();
  __threadfence();   // acquire: invalidate stale near caches before reading
}

// online LSE over one 2048-float row (wave-parallel, 8-wide micro-blocks,
// two b128 loads per iteration => ~1.4 exp/element)
__device__ __forceinline__ void ot_row_lse(const float* __restrict__ zrow,
                                           const float* __restrict__ add,
                                           int lane, float& mx_out, float& s_out) {
  const float4* z4p = (const float4*)zrow;
  const float4* a4p = (const float4*)add;
  float mx = -3.4e38f, s = 0.f;
  for (int base = 0; base < N2048 / 4; base += 64) {
    int q0 = base + lane;
    int q1 = base + 32 + lane;
    float4 za = z4p[q0], zb = z4p[q1];
    float4 aa = a4p[q0], ab = a4p[q1];
    float x0 = za.x + aa.x, x1 = za.y + aa.y, x2 = za.z + aa.z, x3 = za.w + aa.w;
    float x4 = zb.x + ab.x, x5 = zb.y + ab.y, x6 = zb.z + ab.z, x7 = zb.w + ab.w;
    float m8 = fmaxf(fmaxf(fmaxf(x0, x1), fmaxf(x2, x3)),
                     fmaxf(fmaxf(x4, x5), fmaxf(x6, x7)));
    float s8 = __expf(x0 - m8) + __expf(x1 - m8) + __expf(x2 - m8) + __expf(x3 - m8)
             + __expf(x4 - m8) + __expf(x5 - m8) + __expf(x6 - m8) + __expf(x7 - m8);
    float m2 = fmaxf(mx, m8);
    s = s * __expf(mx - m2) + s8 * __expf(m8 - m2);
    mx = m2;
  }
  for (int off = 16; off > 0; off >>= 1) {
    float mo = __shfl_xor(mx, off, 32);
    float so = __shfl_xor(s, off, 32);
    float m2 = fmaxf(mx, mo);
    s = s * __expf(mx - m2) + so * __expf(mo - m2);
    mx = m2;
  }
  mx_out = mx;
  s_out = s;
}

// ------------------------------------ persistent Sinkhorn (single launch)
// 500 iterations; row pass over Z, column pass over ZT; grid barrier between.
__global__ __launch_bounds__(256) void ot_sinkhorn_kernel(
    const float* __restrict__ Z, const float* __restrict__ ZT,
    const float* __restrict__ lmu, const float* __restrict__ lnu,
    float* __restrict__ u, float* __restrict__ v, unsigned* __restrict__ bar) {
  const int lane = threadIdx.x & 31;
  const int gw = blockIdx.x * 8 + (threadIdx.x >> 5);  // global wave id
  unsigned* cnt = bar;
  unsigned* gen = bar + 1;

  for (int it = 0; it < SINKHORN_ITERS; ++it) {
    // u = lmu - LSE_j(Z[row,:] + v)
    for (int row = gw; row < N2048; row += SK_WAVES) {
      float mx, s;
      ot_row_lse(Z + (size_t)row * N2048, v, lane, mx, s);
      if (lane == 0) u[row] = lmu[row] - (mx + __logf(s));
    }
    ot_grid_sync(cnt, gen, SK_BLOCKS);

    // v = lnu - LSE_i(ZT[col,:] + u)
    for (int col = gw; col < N2048; col += SK_WAVES) {
      float mx, s;
      ot_row_lse(ZT + (size_t)col * N2048, u, lane, mx, s);
      if (lane == 0) v[col] = lnu[col] - (mx + __logf(s));
    }
    ot_grid_sync(cnt, gen, SK_BLOCKS);
  }
}

// ------------------------------------------------------------- GEMM 2
// T[i,d] = sum_j exp(Z[i,j] + u[i] + v[j]) * TF[d,j]; A-tiles read from ZT
// (lane index is M, so ZT rows give coalesced A loads).
__global__ __launch_bounds__(256) void ot_gemm2_kernel(
    const float* __restrict__ ZT, const float* __restrict__ u,
    const float* __restrict__ v, const float* __restrict__ TF,
    float* __restrict__ T) {
  const int lane = threadIdx.x & 31;
  const int wave = threadIdx.x >> 5;
  const int i0 = blockIdx.x * 64 + (wave & 1) * 32;    // P rows (M)
  const int d0 = blockIdx.y * 128 + (wave >> 1) * 32;  // feature cols (N)
  const int r16 = lane & 15;
  const int g   = lane >> 4;
  const float u0 = u[i0 + r16];
  const float u1 = u[i0 + 16 + r16];

  v8f acc00 = {}, acc01 = {}, acc10 = {}, acc11 = {};
  for (int j0 = 0; j0 < N2048; j0 += 4) {
    const int k0 = j0 + 2 * g;
    const float vv0 = v[k0];
    const float vv1 = v[k0 + 1];
    v2f a0, a1, b0, b1;
    const float* zc0 = ZT + (size_t)k0 * N2048;        // column k0 of Z
    const float* zc1 = ZT + (size_t)(k0 + 1) * N2048;  // column k0+1 of Z
    a0.x = __expf(zc0[i0 + r16] + u0 + vv0);
    a0.y = __expf(zc1[i0 + r16] + u0 + vv1);
    a1.x = __expf(zc0[i0 + 16 + r16] + u1 + vv0);
    a1.y = __expf(zc1[i0 + 16 + r16] + u1 + vv1);
    const float* t0 = TF + (size_t)(d0 + r16) * N2048 + k0;
    const float* t1 = TF + (size_t)(d0 + 16 + r16) * N2048 + k0;
    b0.x = t0[0];  b0.y = t0[1];
    b1.x = t1[0];  b1.y = t1[1];
    acc00 = wmma_f32_k4(a0, b0, acc00);
    acc01 = wmma_f32_k4(a0, b1, acc01);
    acc10 = wmma_f32_k4(a1, b0, acc10);
    acc11 = wmma_f32_k4(a1, b1, acc11);
  }

#pragma unroll
  for (int r = 0; r < 8; ++r) {
    int ri0 = i0 + 8 * g + r;
    int ri1 = ri0 + 16;
    int cd0 = d0 + r16;
    int cd1 = cd0 + 16;
    T[(size_t)ri0 * N2048 + cd0] = acc00[r];
    T[(size_t)ri0 * N2048 + cd1] = acc01[r];
    T[(size_t)ri1 * N2048 + cd0] = acc10[r];
    T[(size_t)ri1 * N2048 + cd1] = acc11[r];
  }
}

// ------------------------------------------------------- loss reduction
__global__ __launch_bounds__(256) void ot_loss_kernel(
    const float* __restrict__ QF, const float* __restrict__ qa,
    const float* __restrict__ label, const float* __restrict__ T,
    float* __restrict__ out) {
  const int i = blockIdx.x * 256 + threadIdx.x;
  const float a = qa[i];
  float s = 0.f;
  for (int d = 0; d < N2048; ++d) {
    float r = a * QF[(size_t)d * N2048 + i] - T[(size_t)i * N2048 + d];
    s = fmaf(r, r, s);
  }
  float dist = sqrtf(s);
  float l = label[i];
  float hinge = fmaxf(OT_MARGIN - dist, 0.f);
  float loss = 0.5f * l * s + 0.5f * (1.f - l) * hinge * hinge;

  __shared__ float red[256];
  red[threadIdx.x] = loss;
  __syncthreads();
  for (int off = 128; off > 0; off >>= 1) {
    if (threadIdx.x < off) red[threadIdx.x] += red[threadIdx.x + off];
    __syncthreads();
  }
  if (threadIdx.x == 0) atomicAdd(out, red[0]);
}

// ---------------------------------------------------------------- host
extern "C" void kernel_launch(void* const* d_in, const int* in_sizes, int n_in,
                              void* d_out, int out_size, void* d_ws, size_t ws_size,
                              hipStream_t stream) {
  const float* QF    = (const float*)d_in[0];  // query_features  [D, m]
  const float* qa    = (const float*)d_in[1];  // query_attention [m]
  const float* TF    = (const float*)d_in[2];  // target_features [D, n]
  const float* ta    = (const float*)d_in[3];  // target_attention[n]
  const float* label = (const float*)d_in[4];  // label [m]
  float* out = (float*)d_out;

  char* ws = (char*)d_ws;
  float* Z   = (float*)(ws);                       // 16 MB
  float* ZT  = (float*)(ws + ((size_t)16 << 20));  // 16 MB
  float* T   = Z;                                  // aliases Z (dead by GEMM-2)
  float* u   = (float*)(ws + ((size_t)32 << 20));  // 2048
  float* v   = u + N2048;
  float* lmu = v + N2048;
  float* lnu = lmu + N2048;
  float* qsq = lnu + N2048;
  float* tsq = qsq + N2048;
  unsigned* bar = (unsigned*)(tsq + N2048);        // [cnt, gen]
  (void)in_sizes; (void)n_in; (void)out_size; (void)ws_size;

  ot_init_kernel<<<8, 256, 0, stream>>>(qa, ta, u, v, lmu, lnu, out, bar);
  ot_colsq_kernel<<<8, 256, 0, stream>>>(QF, qsq);
  ot_colsq_kernel<<<8, 256, 0, stream>>>(TF, tsq);
  ot_gemm1_kernel<<<dim3(32, 16), 256, 0, stream>>>(QF, TF, qsq, tsq, Z, ZT);

  // one persistent launch: 500 Sinkhorn iterations with grid barriers
  ot_sinkhorn_kernel<<<SK_BLOCKS, 256, 0, stream>>>(Z, ZT, lmu, lnu, u, v, bar);

  ot_gemm2_kernel<<<dim3(32, 16), 256, 0, stream>>>(ZT, u, v, TF, T);
  ot_loss_kernel<<<8, 256, 0, stream>>>(QF, qa, label, T, out);
}